// SSIM_16406775070940
// MI455X (gfx1250) — compile-verified
//
#include <hip/hip_runtime.h>
#include <hip/hip_bf16.h>
#include <math.h>

// SSIM (11x11 Gaussian, sigma=1.5) on 64x1x512x512 fp32 via separable conv
// mapped onto V_WMMA_F32_16X16X4_F32 band matmuls. One wave = one 16x16
// output tile; two WMMA passes (horizontal then vertical) over K=26 (pad 28).
// Band matrix entries come from a 16-entry LDS LUT (Gaussian taps + zero slot)
// instead of per-lane cndmask chains.

typedef float v2f __attribute__((ext_vector_type(2)));
typedef float v8f __attribute__((ext_vector_type(8)));

#define IMG_W 512
#define IMG_H 512
#define OUT_D 502           // 512 - 11 + 1
#define TILES_X 32          // ceil(502/16)
#define TILES_PER_IMG (TILES_X * TILES_X)
#define N_IMG 64
#define N_TILES (N_IMG * TILES_PER_IMG)   // 65536

__device__ __forceinline__ float shx16(float x) {
    return __shfl_xor(x, 16, 32);
}

__global__ __launch_bounds__(256)
void ssim_tile_kernel(const float* __restrict__ yhat,
                      const float* __restrict__ yref,
                      const float* __restrict__ win,
                      float* __restrict__ tile_sums)
{
    const int lane = threadIdx.x & 31;
    const int wave = threadIdx.x >> 5;
    const int tile = blockIdx.x * 8 + wave;
    const int img  = tile >> 10;          // /1024 tiles per image
    const int trem = tile & 1023;
    const int ty   = trem >> 5;
    const int tx   = trem & 31;
    const int row0 = ty * 16;
    const int col0 = tx * 16;

    const float* A0 = yhat + (size_t)img * (IMG_H * IMG_W);
    const float* B0 = yref + (size_t)img * (IMG_H * IMG_W);

    const int cc = lane & 15;     // col (as B/N) or row (as A/M) index
    const int hi = lane >> 4;     // lane half

    // 1D gaussian taps from the 2D window diagonal: w[i] = sqrt(win[i][i]).
    // LDS LUT: slots 0..10 = taps, slot 11..15 = 0 (out-of-band).
    __shared__ float w_lut[16];
    if (threadIdx.x < 16) {
        float v = 0.f;
        if (threadIdx.x < 11) v = sqrtf(win[threadIdx.x * 12]);
        w_lut[threadIdx.x] = v;
    }
    __syncthreads();

    // Band chunks, shared by both passes (identical lane layout):
    //   as B (4x16) for pass1:  B[k][c] = w[k-c]
    //   as A (16x4) for pass2:  A[r][k] = w[k-r]
    // chunk q, element p: k = 4q + p + 2*hi, other index = cc.
    v2f band[7];
#pragma unroll
    for (int q = 0; q < 7; ++q) {
#pragma unroll
        for (int p = 0; p < 2; ++p) {
            const int k = 4 * q + p + 2 * hi;
            const int d = k - cc;
            const bool inband = ((unsigned)d <= 10u) && (k < 26);
            const int idx = inband ? d : 11;
            band[q][p] = w_lut[idx];
        }
    }

    const v8f zero8 = {0.f, 0.f, 0.f, 0.f, 0.f, 0.f, 0.f, 0.f};

    // ---------- Pass 1: horizontal conv, tmp(32x16) per map ----------
    v8f tacc[5][2];
#pragma unroll
    for (int m = 0; m < 5; ++m) { tacc[m][0] = zero8; tacc[m][1] = zero8; }

#pragma unroll
    for (int q = 0; q < 7; ++q) {
        const int k0 = 4 * q;
        int col = col0 + k0 + 2 * hi;
        col = col > 510 ? 510 : col;              // keep float2 in-bounds
#pragma unroll
        for (int t = 0; t < 2; ++t) {
            int row = row0 + t * 16 + cc;
            row = row > 511 ? 511 : row;          // clamp (masked later)
            const float* pa = A0 + row * IMG_W + col;
            const float* pb = B0 + row * IMG_W + col;
            v2f va = *(const v2f*)pa;
            v2f vb = *(const v2f*)pb;
            v2f vaa = va * va;
            v2f vbb = vb * vb;
            v2f vab = va * vb;
            tacc[0][t] = __builtin_amdgcn_wmma_f32_16x16x4_f32(false, va,  false, band[q], (short)0, tacc[0][t], false, false);
            tacc[1][t] = __builtin_amdgcn_wmma_f32_16x16x4_f32(false, vb,  false, band[q], (short)0, tacc[1][t], false, false);
            tacc[2][t] = __builtin_amdgcn_wmma_f32_16x16x4_f32(false, vaa, false, band[q], (short)0, tacc[2][t], false, false);
            tacc[3][t] = __builtin_amdgcn_wmma_f32_16x16x4_f32(false, vbb, false, band[q], (short)0, tacc[3][t], false, false);
            tacc[4][t] = __builtin_amdgcn_wmma_f32_16x16x4_f32(false, vab, false, band[q], (short)0, tacc[4][t], false, false);
        }
    }

    // ---------- Pass 2: vertical conv, out(16x16) per map ----------
    // Reshape tmp C/D layout (rows m / m+8 per VGPR) into B layout
    // (rows k0 / k0+2 per VGPR) via half-wave swaps.
    const bool hiHalf = (lane & 16) != 0;
    v8f oacc[5];
#pragma unroll
    for (int m = 0; m < 5; ++m) oacc[m] = zero8;

#pragma unroll
    for (int m = 0; m < 5; ++m) {
#pragma unroll
        for (int q = 0; q < 7; ++q) {
            v2f Bc;
#pragma unroll
            for (int p = 0; p < 2; ++p) {
                const int rlo = 4 * q + p;        // tmp row for lanes 0-15
                const int tl  = rlo >> 4;         // tmp row-tile
                const int jj  = rlo & 15;
                const int v   = jj & 7;           // source VGPR (row rlo)
                const int h   = jj >> 3;          // source half
                float xlo = tacc[m][tl][v];       // holds rows (tl*16+v, +8)
                float xhi = tacc[m][tl][v + 2];   // holds rows (tl*16+v+2, +8)
                float lo = h ? shx16(xlo) : xlo;          // row rlo -> lanes 0-15
                float hh = h ? xhi       : shx16(xhi);    // row rlo+2 -> lanes 16-31
                Bc[p] = hiHalf ? hh : lo;
            }
            oacc[m] = __builtin_amdgcn_wmma_f32_16x16x4_f32(false, band[q], false, Bc, (short)0, oacc[m], false, false);
        }
    }

    // ---------- Pass 3: SSIM formula + masked tile sum ----------
    const float C1 = 1e-4f;   // (0.01*1)^2
    const float C2 = 9e-4f;   // (0.03*1)^2
    float lsum = 0.f;
#pragma unroll
    for (int v = 0; v < 8; ++v) {
        const float mu1 = oacc[0][v], mu2 = oacc[1][v];
        const float s11 = oacc[2][v], s22 = oacc[3][v], s12 = oacc[4][v];
        const float mu1sq = mu1 * mu1;
        const float mu2sq = mu2 * mu2;
        const float mu12  = mu1 * mu2;
        const float num = (2.f * mu12 + C1) * (2.f * (s12 - mu12) + C2);
        const float den = (mu1sq + mu2sq + C1) *
                          ((s11 - mu1sq) + (s22 - mu2sq) + C2);
        const int grow = row0 + v + 8 * hi;
        const int gcol = col0 + cc;
        const bool valid = (grow < OUT_D) && (gcol < OUT_D);
        lsum += valid ? (num / den) : 0.f;
    }
#pragma unroll
    for (int s = 16; s > 0; s >>= 1) lsum += __shfl_xor(lsum, s, 32);
    if (lane == 0) tile_sums[tile] = lsum;
}

// Deterministic final reduction: fixed per-thread strided order + LDS tree.
__global__ __launch_bounds__(256)
void ssim_reduce_kernel(const float* __restrict__ tile_sums,
                        float* __restrict__ out)
{
    __shared__ float sm[256];
    float acc = 0.f;
    for (int i = threadIdx.x; i < N_TILES; i += 256) acc += tile_sums[i];
    sm[threadIdx.x] = acc;
    __syncthreads();
    for (int s = 128; s > 0; s >>= 1) {
        if (threadIdx.x < s) sm[threadIdx.x] += sm[threadIdx.x + s];
        __syncthreads();
    }
    if (threadIdx.x == 0)
        out[0] = sm[0] * (1.0f / ((float)N_IMG * (float)OUT_D * (float)OUT_D));
}

extern "C" void kernel_launch(void* const* d_in, const int* in_sizes, int n_in,
                              void* d_out, int out_size, void* d_ws, size_t ws_size,
                              hipStream_t stream) {
    const float* yhat = (const float*)d_in[0];
    const float* yref = (const float*)d_in[1];
    const float* win  = (const float*)d_in[2];
    float* sums = (float*)d_ws;           // N_TILES floats = 256 KB
    float* outp = (float*)d_out;

    ssim_tile_kernel<<<N_TILES / 8, 256, 0, stream>>>(yhat, yref, win, sums);
    ssim_reduce_kernel<<<1, 256, 0, stream>>>(sums, outp);
}